// GenerateTFModule_55284819034743
// MI455X (gfx1250) — compile-verified
//
#include <hip/hip_runtime.h>

// Problem dims (fixed by the reference)
constexpr int kB  = 8;
constexpr int kN  = 512;
constexpr int kNN = kN * kN;          // 262144 elements per batch
constexpr int kT  = 1024;             // bins
constexpr int kF  = 8;                // weight features
constexpr int kHS = 9;                // padded hist stride (LDS bank spread)

#define THREADS 256
constexpr int kBlocksPerBatch = 64;   // 64*256*16 = 262144
constexpr int kEPT = 16;              // elements per thread

// ---------------------------------------------------------------------------
// Kernel 0: zero the output and init per-batch argmin keys / below-bin0 counts
// ---------------------------------------------------------------------------
__global__ __launch_bounds__(THREADS) void init_kernel(
    float* __restrict__ out,
    unsigned long long* __restrict__ minkey,
    unsigned int* __restrict__ cnt0) {
  int i = blockIdx.x * THREADS + threadIdx.x;
  if (i < kB * kT * kF) out[i] = 0.0f;
  if (i < kB) { minkey[i] = ~0ull; cnt0[i] = 0u; }
}

// ---------------------------------------------------------------------------
// Kernel 1: main streaming histogram
//   grid = (kBlocksPerBatch, kB); each block owns 4096 elements of one batch.
//   LDS: 1024x8 f32 histogram (stride 9) + 1024 f32 bin edges (async-copied).
// ---------------------------------------------------------------------------
__global__ __launch_bounds__(THREADS) void hist_kernel(
    const float* __restrict__ input,
    const float* __restrict__ bins,
    const float* __restrict__ weights,
    float* __restrict__ out,
    unsigned long long* __restrict__ minkey,
    unsigned int* __restrict__ cnt0) {
  __shared__ float sh_hist[kT * kHS];     // 36864 B
  __shared__ float sh_bins[kT];           //  4096 B
  __shared__ unsigned long long sh_minkey;
  __shared__ unsigned int sh_cnt0;

  const int tid   = threadIdx.x;
  const int b     = blockIdx.y;
  const int chunk = blockIdx.x;

  for (int i = tid; i < kT * kHS; i += THREADS) sh_hist[i] = 0.0f;
  if (tid == 0) { sh_minkey = ~0ull; sh_cnt0 = 0u; }

  // ---- CDNA5 async global->LDS copy of this batch's 4 KB bin table --------
  // GLOBAL_LOAD_ASYNC_TO_LDS_B128, GVS mode: saddr base + 32-bit vgpr offset.
  {
    unsigned lds_addr = (unsigned)(size_t)(&sh_bins[tid * 4]);   // LDS byte offset
    unsigned goff     = (unsigned)(tid * 16);                    // 16 B per lane
    unsigned long long gbase = (unsigned long long)(size_t)(bins + (size_t)b * kT);
    asm volatile("global_load_async_to_lds_b128 %0, %1, %2 offset:0"
                 :: "v"(lds_addr), "v"(goff), "s"(gbase) : "memory");
    asm volatile("s_wait_asynccnt 0x0" ::: "memory");
  }
  __syncthreads();

  // ---- stream 16 contiguous elements per thread ---------------------------
  const int e0 = chunk * (THREADS * kEPT) + tid * kEPT;          // index in batch
  const float4* vin = (const float4*)(input + (size_t)b * kNN + e0);

  float vmin = 3.402823466e+38f;
  int imin = 0;
  unsigned c0 = 0;

#pragma unroll
  for (int g = 0; g < 4; ++g) {
    float4 q = vin[g];
    float vv[4] = {q.x, q.y, q.z, q.w};
#pragma unroll
    for (int u = 0; u < 4; ++u) {
      float v = vv[u];
      int e = e0 + g * 4 + u;
      if (v < vmin) { vmin = v; imin = e; }

      // j = number of bin edges <= v  (branchless binary search in LDS)
      int j;
      if (sh_bins[kT - 1] <= v) {
        j = kT;                          // above last edge: contributes nowhere
      } else {
        j = 0;
#pragma unroll
        for (int s = 512; s >= 1; s >>= 1)
          j = (sh_bins[j + s - 1] <= v) ? (j + s) : j;
      }
      c0 += (j == 0) ? 1u : 0u;

      if (j < kT) {
        const float* wp = weights + ((size_t)b * kNN + (size_t)e) * kF;
        float4 w0 = *(const float4*)(wp);
        float4 w1 = *(const float4*)(wp + 4);
        float* h = &sh_hist[j * kHS];
        atomicAdd(h + 0, w0.x); atomicAdd(h + 1, w0.y);
        atomicAdd(h + 2, w0.z); atomicAdd(h + 3, w0.w);
        atomicAdd(h + 4, w1.x); atomicAdd(h + 5, w1.y);
        atomicAdd(h + 6, w1.z); atomicAdd(h + 7, w1.w);
      }
    }
  }

  // ---- per-block reductions: argmin key (order-preserving u64) + count0 ---
  unsigned kb = __float_as_uint(vmin);
  kb = kb ^ ((kb & 0x80000000u) ? 0xFFFFFFFFu : 0x80000000u);
  unsigned long long key =
      ((unsigned long long)kb << 32) | (unsigned long long)(unsigned)imin;
  atomicMin(&sh_minkey, key);
  if (c0) atomicAdd(&sh_cnt0, c0);
  __syncthreads();

  if (tid == 0) {
    atomicMin(&minkey[b], sh_minkey);
    if (sh_cnt0) atomicAdd(&cnt0[b], sh_cnt0);
  }

  // ---- flush LDS histogram to global with f32 atomics ---------------------
  for (int i = tid; i < kT * kF; i += THREADS) {
    int j = i >> 3, f = i & 7;
    float v = sh_hist[j * kHS + f];
    if (v != 0.0f)
      unsafeAtomicAdd(&out[((size_t)b * kT + j) * kF + f], v);
  }
}

// ---------------------------------------------------------------------------
// Kernel 2: replicate the reference's clip(counts-1, 0) quirk:
// if NO element falls below bins[b,0], out[b,0,:] = weights of the min element.
// ---------------------------------------------------------------------------
__global__ void fixup_kernel(const float* __restrict__ weights,
                             const unsigned long long* __restrict__ minkey,
                             const unsigned int* __restrict__ cnt0,
                             float* __restrict__ out) {
  int t = threadIdx.x;
  if (t >= kB * kF) return;
  int b = t >> 3, f = t & 7;
  if (cnt0[b] == 0u) {
    unsigned i = (unsigned)(minkey[b] & 0xFFFFFFFFull);
    out[(size_t)b * kT * kF + f] = weights[((size_t)b * kNN + i) * kF + f];
  }
}

// ---------------------------------------------------------------------------
extern "C" void kernel_launch(void* const* d_in, const int* in_sizes, int n_in,
                              void* d_out, int out_size, void* d_ws, size_t ws_size,
                              hipStream_t stream) {
  const float* input   = (const float*)d_in[0];  // [B,N,N] f32
  const float* bins    = (const float*)d_in[1];  // [B,T]   f32 sorted
  const float* weights = (const float*)d_in[2];  // [B,N,N,F] f32
  float* out = (float*)d_out;                    // [B,T,F] f32

  unsigned long long* minkey = (unsigned long long*)d_ws;            // 8 * u64
  unsigned int* cnt0 = (unsigned int*)((char*)d_ws + 64);            // 8 * u32

  init_kernel<<<(kB * kT * kF + THREADS - 1) / THREADS, THREADS, 0, stream>>>(
      out, minkey, cnt0);

  dim3 grid(kBlocksPerBatch, kB);
  hist_kernel<<<grid, THREADS, 0, stream>>>(input, bins, weights, out, minkey, cnt0);

  fixup_kernel<<<1, 64, 0, stream>>>(weights, minkey, cnt0, out);
}